// Attn_55611236548746
// MI455X (gfx1250) — compile-verified
//
#include <hip/hip_runtime.h>
#include <math.h>

// ---------------------------------------------------------------------------
// Attention pooling, MI455X (gfx1250, wave32).
// Memory-bound: enc (256 MB) streamed twice -> ~22us floor @ 23.3 TB/s.
// f32 WMMA 16x16x4 used as the accumulate engine on both passes.
// ---------------------------------------------------------------------------

typedef __attribute__((ext_vector_type(2))) float v2f;
typedef __attribute__((ext_vector_type(8))) float v8f;

#define BATCH   32
#define TLEN    2048
#define HEDIM   1024
#define NCHUNK  4          // t-dimension split for the weighted-sum pass

// ---------------------------------------------------------------------------
// Pass 1: raw energies[b,t] = dot(enc[b,t,:], w_e)
// One wave per 16 t-rows. A(16x4) = enc rows (ISA f32 A layout: lanes 0-15
// hold K=0..1 in v[0..1], lanes 16-31 hold K=2..3). B(4x16) = w_e[e+k]
// broadcast across N (B VGPR v, lanes<16 -> K=v, lanes>=16 -> K=v+2).
// D columns are identical; D[m,0] = energy(t0+m).
// ---------------------------------------------------------------------------
__global__ void __launch_bounds__(256) k_energies(const float* __restrict__ enc,
                                                  const float* __restrict__ w_e,
                                                  float* __restrict__ energies) {
    const int lane = threadIdx.x & 31;
    const int wave = threadIdx.x >> 5;
    const int b    = blockIdx.x;                   // 0..31
    const int t0   = (blockIdx.y * 8 + wave) * 16; // 16 t-rows per wave
    const int m    = lane & 15;
    const int hi   = lane >> 4;                    // 0: K=0..1, 1: K=2..3

    const float* rowE = enc + ((size_t)b * TLEN + (size_t)(t0 + m)) * HEDIM + hi * 2;
    const float* wp   = w_e + hi * 2;

    v8f acc0 = {};
    v8f acc1 = {};
    for (int e = 0; e < HEDIM; e += 8) {
        v2f a0 = *(const v2f*)(rowE + e);
        v2f b0 = *(const v2f*)(wp   + e);
        v2f a1 = *(const v2f*)(rowE + e + 4);
        v2f b1 = *(const v2f*)(wp   + e + 4);
        acc0 = __builtin_amdgcn_wmma_f32_16x16x4_f32(false, a0, false, b0,
                                                     (short)0, acc0, false, false);
        acc1 = __builtin_amdgcn_wmma_f32_16x16x4_f32(false, a1, false, b1,
                                                     (short)0, acc1, false, false);
        __builtin_prefetch(rowE + e + 256, 0, 1);  // global_prefetch_b8, 1KB ahead
    }
    v8f acc = acc0 + acc1;

    // lane 0 holds D[0..7, 0] (M=0..7), lane 16 holds D[8..15, 0] (M=8..15)
    if ((lane & 15) == 0) {
        float* out = energies + (size_t)b * TLEN + t0 + hi * 8;
#pragma unroll
        for (int v = 0; v < 8; ++v) out[v] = acc[v];
    }
}

// ---------------------------------------------------------------------------
// Pass 2: per-batch scalar projection + masked softmax (tiny; one block/batch)
// ---------------------------------------------------------------------------
__global__ void __launch_bounds__(256) k_softmax(const float* __restrict__ energies,
                                                 const float* __restrict__ hidden,   // (2,32,1024)
                                                 const float* __restrict__ mask,     // (32,2048)
                                                 const float* __restrict__ attn_w,   // w_h = [0:2048)
                                                 const float* __restrict__ attn_b,
                                                 float* __restrict__ probs) {
    __shared__ float red[256];
    __shared__ float e_sh[TLEN];
    const int b   = blockIdx.x;
    const int tid = threadIdx.x;

    // s_b = dot(hid_flat[b], w_h) + bias ; hid_flat[b][j] = hidden[j/1024, b, j%1024]
    float part = 0.f;
    for (int j = tid; j < 2048; j += 256) {
        float h = (j < 1024) ? hidden[(size_t)b * 1024 + j]
                             : hidden[(size_t)32 * 1024 + (size_t)b * 1024 + (j - 1024)];
        part += h * attn_w[j];
    }
    red[tid] = part; __syncthreads();
    for (int s = 128; s > 0; s >>= 1) { if (tid < s) red[tid] += red[tid + s]; __syncthreads(); }
    const float s_b = red[0] + attn_b[0];
    __syncthreads();

    // e = (raw + s_b) * mask ; max
    float mymax = -INFINITY;
    for (int t = tid; t < TLEN; t += 256) {
        float e = (energies[(size_t)b * TLEN + t] + s_b) * mask[(size_t)b * TLEN + t];
        e_sh[t] = e;
        mymax = fmaxf(mymax, e);
    }
    red[tid] = mymax; __syncthreads();
    for (int s = 128; s > 0; s >>= 1) { if (tid < s) red[tid] = fmaxf(red[tid], red[tid + s]); __syncthreads(); }
    const float mx = red[0];
    __syncthreads();

    // exp / sum
    float mysum = 0.f;
    for (int t = tid; t < TLEN; t += 256) {
        float ex = __expf(e_sh[t] - mx);
        e_sh[t] = ex;
        mysum += ex;
    }
    red[tid] = mysum; __syncthreads();
    for (int s = 128; s > 0; s >>= 1) { if (tid < s) red[tid] += red[tid + s]; __syncthreads(); }
    const float denom = red[0];
    __syncthreads();

    // p = softmax * mask, then renormalize by masked sum
    float msum = 0.f;
    for (int t = tid; t < TLEN; t += 256) {
        float p = (e_sh[t] / denom) * mask[(size_t)b * TLEN + t];
        e_sh[t] = p;
        msum += p;
    }
    red[tid] = msum; __syncthreads();
    for (int s = 128; s > 0; s >>= 1) { if (tid < s) red[tid] += red[tid + s]; __syncthreads(); }
    const float norm = red[0];
    __syncthreads();

    for (int t = tid; t < TLEN; t += 256)
        probs[(size_t)b * TLEN + t] = e_sh[t] / norm;
}

// ---------------------------------------------------------------------------
// Pass 3: out[b,e] = sum_t p[b,t] * enc[b,t,e]
// A(16x4) = p[t+k] broadcast over M; B(4x16) = enc[t+k, e0..e0+15].
// D rows identical; D[0, n] at lanes 0-15 = out[b, e0+n]. t split into
// NCHUNK chunks (deterministic partials, no float atomics).
// ---------------------------------------------------------------------------
__global__ void __launch_bounds__(256) k_wsum(const float* __restrict__ enc,
                                              const float* __restrict__ probs,
                                              float* __restrict__ partials) { // [NCHUNK][32][1024]
    const int lane = threadIdx.x & 31;
    const int wave = threadIdx.x >> 5;
    const int b    = blockIdx.x;                     // 0..31
    const int e0   = (blockIdx.y * 8 + wave) * 16;   // 64 e-tiles of 16
    const int chunk = blockIdx.z;                    // 0..NCHUNK-1
    const int hi   = lane >> 4;
    const int col  = e0 + (lane & 15);

    const int tBeg = chunk * (TLEN / NCHUNK);
    const int tEnd = tBeg + (TLEN / NCHUNK);

    const float* pB = probs + (size_t)b * TLEN;
    const float* eB = enc + (size_t)b * TLEN * HEDIM;

    v8f acc0 = {};
    v8f acc1 = {};
    for (int t = tBeg; t < tEnd; t += 8) {
        const int k0 = t + hi * 2;
        v2f a0 = { pB[k0], pB[k0 + 1] };
        v2f b0 = { eB[(size_t)k0 * HEDIM + col], eB[(size_t)(k0 + 1) * HEDIM + col] };
        acc0 = __builtin_amdgcn_wmma_f32_16x16x4_f32(false, a0, false, b0,
                                                     (short)0, acc0, false, false);
        const int k1 = t + 4 + hi * 2;
        v2f a1 = { pB[k1], pB[k1 + 1] };
        v2f b1 = { eB[(size_t)k1 * HEDIM + col], eB[(size_t)(k1 + 1) * HEDIM + col] };
        acc1 = __builtin_amdgcn_wmma_f32_16x16x4_f32(false, a1, false, b1,
                                                     (short)0, acc1, false, false);
        __builtin_prefetch(eB + (size_t)(t + 32) * HEDIM + col, 0, 1);
    }
    v8f acc = acc0 + acc1;

    if (hi == 0) {  // lanes 0-15: D[0, n] = partial out for column e0+n
        partials[((size_t)chunk * BATCH + b) * HEDIM + col] = acc[0];
    }
}

// ---------------------------------------------------------------------------
// Pass 4: reduce the NCHUNK partials into d_out (fixed order -> deterministic)
// ---------------------------------------------------------------------------
__global__ void __launch_bounds__(256) k_reduce(const float* __restrict__ partials,
                                                float* __restrict__ out) {
    const int i = blockIdx.x * 256 + threadIdx.x;   // 0 .. 32*1024-1
    float s = 0.f;
#pragma unroll
    for (int c = 0; c < NCHUNK; ++c)
        s += partials[(size_t)c * BATCH * HEDIM + i];
    out[i] = s;
}

// ---------------------------------------------------------------------------
extern "C" void kernel_launch(void* const* d_in, const int* in_sizes, int n_in,
                              void* d_out, int out_size, void* d_ws, size_t ws_size,
                              hipStream_t stream) {
    const float* hidden = (const float*)d_in[0];   // (2,32,1024)
    const float* enc    = (const float*)d_in[1];   // (32,2048,1024)
    const float* mask   = (const float*)d_in[2];   // (32,2048)
    const float* attn_w = (const float*)d_in[3];   // (3072,)
    const float* attn_b = (const float*)d_in[4];   // (1,)
    float* out = (float*)d_out;                    // (32,1024)

    // workspace: energies (256KB) | probs (256KB) | partials (512KB) = 1MB
    float* energies = (float*)d_ws;
    float* probs    = energies + (size_t)BATCH * TLEN;
    float* partials = probs    + (size_t)BATCH * TLEN;

    dim3 g1(BATCH, TLEN / (16 * 8));               // 32 x 16 blocks, 8 waves each
    k_energies<<<g1, 256, 0, stream>>>(enc, attn_w + 2048, energies);

    k_softmax<<<BATCH, 256, 0, stream>>>(energies, hidden, mask, attn_w, attn_b, probs);

    dim3 g3(BATCH, HEDIM / (16 * 8), NCHUNK);      // 32 x 8 x 4 blocks, 8 waves each
    k_wsum<<<g3, 256, 0, stream>>>(enc, probs, partials);

    k_reduce<<<(BATCH * HEDIM) / 256, 256, 0, stream>>>(partials, out);
}